// BoxFeatureExtractor_7370163880508
// MI455X (gfx1250) — compile-verified
//
#include <hip/hip_runtime.h>
#include <math.h>

#define GS 6
#define NCELL 216
#define MLP_WAVES 2
#define TILES_PER_WAVE 8

typedef __attribute__((ext_vector_type(2))) float v2f;
typedef __attribute__((ext_vector_type(8))) float v8f;

// ---------------- fp32 WMMA helpers (V_WMMA_F32_16X16X4_F32) ----------------
// A 16x4 layout: lane(half,m): v0 = A[m][k0+2*half], v1 = A[m][k0+2*half+1]
// B 4x16 layout: lane(half,m): v0 = B[k0+2*half][n], v1 = B[k0+2*half+1][n]
// C/D 16x16:     lane(half,m): acc[r] = D[r+8*half][m]
__device__ __forceinline__ v8f wmma4(v2f a, v2f b, v8f c) {
  return __builtin_amdgcn_wmma_f32_16x16x4_f32(false, a, false, b, (short)0, c,
                                               false, false);
}

__device__ __forceinline__ v2f frag_a(const float* A, int lda, int m, int half,
                                      int k0) {
  int k = k0 + 2 * half;
  v2f a;
  a.x = A[m * lda + k];
  a.y = A[m * lda + k + 1];
  return a;
}

// B stored paired along K: Bp[k2][n] = {B[2*k2][n], B[2*k2+1][n]} -> one b64
__device__ __forceinline__ v2f frag_bp(const float2* Bp, int ldn, int m,
                                       int half, int k0, int n0) {
  int k2 = (k0 >> 1) + half;
  float2 t = Bp[k2 * ldn + n0 + m];
  v2f b;
  b.x = t.x;
  b.y = t.y;
  return b;
}

// 16xK @ KxN-tile(16 cols starting at n0), K multiple of 4
__device__ __forceinline__ v8f mm16(const float* A, int lda, const float2* Bp,
                                    int ldn, int K, int n0, int m, int half) {
  v8f acc = {0.f, 0.f, 0.f, 0.f, 0.f, 0.f, 0.f, 0.f};
  for (int k0 = 0; k0 < K; k0 += 4)
    acc = wmma4(frag_a(A, lda, m, half, k0), frag_bp(Bp, ldn, m, half, k0, n0),
                acc);
  return acc;
}

// ---------------- kernel 0: zero scratch ----------------
__global__ void bfe_zero_kernel(float* __restrict__ p, size_t n) {
  size_t i = (size_t)blockIdx.x * blockDim.x + threadIdx.x;
  if (i < n) p[i] = 0.f;
}

// ---------------- kernel 1: points-in-boxes + voxelize + centroid stats ----
__global__ __launch_bounds__(256) void bfe_pib_kernel(
    const float* __restrict__ coords, const float* __restrict__ boxes,
    float* __restrict__ nxyz, int* __restrict__ seg,
    float* __restrict__ segstat, int P, int N) {
  // bx[n] = {batch, cx, cy, cz, hx, hy, hz, cos(yaw), sin(yaw)}
  // hx = 0.75*dim_x (enlarged-by-1.5 half extent); enlarged dim = 2*hx
  __shared__ float bx[256][10];
  for (int i = threadIdx.x; i < N; i += blockDim.x) {
    const float* b = boxes + (size_t)i * 8;
    float yaw = b[7];
    bx[i][0] = b[0];
    bx[i][1] = b[1];
    bx[i][2] = b[2];
    bx[i][3] = b[3];
    bx[i][4] = b[4] * 0.75f;
    bx[i][5] = b[5] * 0.75f;
    bx[i][6] = b[6] * 0.75f;
    bx[i][7] = cosf(yaw);
    bx[i][8] = sinf(yaw);
    bx[i][9] = 0.f;
  }
  __syncthreads();
  int p = blockIdx.x * 256 + threadIdx.x;
  if (p >= P) return;
  float4 cd = *(const float4*)(coords + (size_t)p * 4);
  int pb = (int)cd.x;
  int idx = -1;
  for (int n = 0; n < N; n++) {
    if ((int)bx[n][0] != pb) continue;
    float dx = cd.y - bx[n][1];
    float dy = cd.z - bx[n][2];
    float dz = cd.w - bx[n][3];
    float c = bx[n][7], s = bx[n][8];
    float lx = dx * c + dy * s;
    float ly = -dx * s + dy * c;
    if (fabsf(lx) < bx[n][4] && fabsf(ly) < bx[n][5] && fabsf(dz) < bx[n][6]) {
      idx = n;
      break;  // first containing box (argmax of bool)
    }
  }
  bool valid = idx >= 0;
  int mi = valid ? idx : 0;  // reference: argmax(all-false)==0
  float dx = cd.y - bx[mi][1];
  float dy = cd.z - bx[mi][2];
  float dz = cd.w - bx[mi][3];
  float c = bx[mi][7], s = bx[mi][8];
  float nx = dx * c + dy * s;  // rotate by -yaw
  float ny = -dx * s + dy * c;
  float nz = dz;
  // voxel coords: tf = clip(new_xyz / enlarged_dims * GS + 3, 0, GS-1e-4)
  float hi = (float)GS - 1e-4f;
  float tfx = fminf(fmaxf(nx / (2.f * bx[mi][4]) * (float)GS + 3.f, 0.f), hi);
  float tfy = fminf(fmaxf(ny / (2.f * bx[mi][5]) * (float)GS + 3.f, 0.f), hi);
  float tfz = fminf(fmaxf(nz / (2.f * bx[mi][6]) * (float)GS + 3.f, 0.f), hi);
  int gx = (int)floorf(tfx);
  int gy = (int)floorf(tfy);
  int gz = (int)floorf(tfz);
  int cell = (gx * GS + gy) * GS + gz;
  int sgi = valid ? mi * NCELL + cell : N * NCELL;
  seg[p] = sgi;
  float4 o;
  o.x = nx;
  o.y = ny;
  o.z = nz;
  o.w = 0.f;
  *(float4*)(nxyz + (size_t)p * 4) = o;
  if (valid) {
    atomicAdd(&segstat[(size_t)sgi * 4 + 0], nx);
    atomicAdd(&segstat[(size_t)sgi * 4 + 1], ny);
    atomicAdd(&segstat[(size_t)sgi * 4 + 2], nz);
    atomicAdd(&segstat[(size_t)sgi * 4 + 3], 1.f);
  }
}

// ---------------- kernel 2: per-point MLPs via fp32 WMMA + voxel max -------
__global__ __launch_bounds__(64) void bfe_mlp_kernel(
    const float* __restrict__ features, const float* __restrict__ nxyz,
    const int* __restrict__ seg, const float* __restrict__ segstat,
    const float* __restrict__ pe_w1, const float* __restrict__ pe_b1,
    const float* __restrict__ pe_w2, const float* __restrict__ pe_b2,
    const float* __restrict__ proj_w, const float* __restrict__ proj_b,
    const float* __restrict__ fc_w, const float* __restrict__ fc_b,
    unsigned int* __restrict__ vox, int P, int nTiles, int NVOX) {
  // weights in paired-K layout: one ds_load_b64 per B fragment
  __shared__ float2 w_pe1p[2][32];    // K padded 3->4, pair (2,3): row 3 zero
  __shared__ float2 w_pe2p[16][32];   // K=32
  __shared__ float2 w_projp[32][32];  // K=64
  __shared__ float2 w_fcp[32][64];    // K=64, N=64
  __shared__ float b_pe1[32], b_pe2[32], b_proj[32], b_fc[64];
  __shared__ float featA[MLP_WAVES][16][68];  // padded to dodge bank conflicts
  __shared__ float xyzA[MLP_WAVES][16][4];
  __shared__ float t1A[MLP_WAVES][16][36];
  __shared__ float catA[MLP_WAVES][16][68];  // [feat_e | pos_e]
  __shared__ int segA[MLP_WAVES][16];

  const int tid = threadIdx.x;
  for (int i = tid; i < 2 * 32; i += 64) {
    int k2 = i >> 5, n = i & 31;
    float2 t;
    t.x = (2 * k2 + 0 < 3) ? pe_w1[(2 * k2 + 0) * 32 + n] : 0.f;
    t.y = (2 * k2 + 1 < 3) ? pe_w1[(2 * k2 + 1) * 32 + n] : 0.f;
    w_pe1p[k2][n] = t;
  }
  for (int i = tid; i < 16 * 32; i += 64) {
    int k2 = i >> 5, n = i & 31;
    float2 t;
    t.x = pe_w2[(2 * k2 + 0) * 32 + n];
    t.y = pe_w2[(2 * k2 + 1) * 32 + n];
    w_pe2p[k2][n] = t;
  }
  for (int i = tid; i < 32 * 32; i += 64) {
    int k2 = i >> 5, n = i & 31;
    float2 t;
    t.x = proj_w[(2 * k2 + 0) * 32 + n];
    t.y = proj_w[(2 * k2 + 1) * 32 + n];
    w_projp[k2][n] = t;
  }
  for (int i = tid; i < 32 * 64; i += 64) {
    int k2 = i >> 6, n = i & 63;
    float2 t;
    t.x = fc_w[(2 * k2 + 0) * 64 + n];
    t.y = fc_w[(2 * k2 + 1) * 64 + n];
    w_fcp[k2][n] = t;
  }
  if (tid < 32) {
    b_pe1[tid] = pe_b1[tid];
    b_pe2[tid] = pe_b2[tid];
    b_proj[tid] = proj_b[tid];
  }
  b_fc[tid] = fc_b[tid];
  __syncthreads();

  const int wave = tid >> 5;
  const int lane = tid & 31;
  const int half = lane >> 4;
  const int m = lane & 15;

  for (int t = 0; t < TILES_PER_WAVE; t++) {
    // tile>=nTiles -> all p>=P, guards below keep this harmless
    const int tile =
        blockIdx.x * (MLP_WAVES * TILES_PER_WAVE) + t * MLP_WAVES + wave;

    // --- stage features (16 points x 64 ch) ---
    {
      int r = lane >> 1;
      int c0 = (lane & 1) * 32;
      int p = tile * 16 + r;
      if (p < P) {
        const float4* src = (const float4*)(features + (size_t)p * 64 + c0);
#pragma unroll
        for (int q = 0; q < 8; q++) {
          float4 v = src[q];
          featA[wave][r][c0 + q * 4 + 0] = v.x;
          featA[wave][r][c0 + q * 4 + 1] = v.y;
          featA[wave][r][c0 + q * 4 + 2] = v.z;
          featA[wave][r][c0 + q * 4 + 3] = v.w;
        }
      } else {
        for (int q = 0; q < 32; q++) featA[wave][r][c0 + q] = 0.f;
      }
    }
    // --- stage centroid-normalized coords + seg ---
    if (lane < 16) {
      int p = tile * 16 + lane;
      if (p < P) {
        int s = seg[p];
        segA[wave][lane] = s;
        float sx = segstat[(size_t)s * 4 + 0];
        float sy = segstat[(size_t)s * 4 + 1];
        float sz = segstat[(size_t)s * 4 + 2];
        float cw = segstat[(size_t)s * 4 + 3];
        float inv = 1.0f / fmaxf(cw, 1.0f);
        xyzA[wave][lane][0] = nxyz[(size_t)p * 4 + 0] - sx * inv;
        xyzA[wave][lane][1] = nxyz[(size_t)p * 4 + 1] - sy * inv;
        xyzA[wave][lane][2] = nxyz[(size_t)p * 4 + 2] - sz * inv;
        xyzA[wave][lane][3] = 0.f;
      } else {
        segA[wave][lane] = 0x7fffffff;
        xyzA[wave][lane][0] = 0.f;
        xyzA[wave][lane][1] = 0.f;
        xyzA[wave][lane][2] = 0.f;
        xyzA[wave][lane][3] = 0.f;
      }
    }
    __syncthreads();

    // --- feat_e = relu(features @ proj_w + b)  ->  catA[:,0:32] ---
    for (int nt = 0; nt < 2; nt++) {
      v8f acc =
          mm16(&featA[wave][0][0], 68, &w_projp[0][0], 32, 64, nt * 16, m, half);
      float bv = b_proj[nt * 16 + m];
#pragma unroll
      for (int r = 0; r < 8; r++) {
        float v = acc[r] + bv;
        catA[wave][r + 8 * half][nt * 16 + m] = v > 0.f ? v : 0.f;
      }
    }
    // --- pos layer 1: relu(norm_xyz @ pe_w1 + b) -> t1A ---
    for (int nt = 0; nt < 2; nt++) {
      v8f acc =
          mm16(&xyzA[wave][0][0], 4, &w_pe1p[0][0], 32, 4, nt * 16, m, half);
      float bv = b_pe1[nt * 16 + m];
#pragma unroll
      for (int r = 0; r < 8; r++) {
        float v = acc[r] + bv;
        t1A[wave][r + 8 * half][nt * 16 + m] = v > 0.f ? v : 0.f;
      }
    }
    __syncthreads();
    // --- pos layer 2 -> catA[:,32:64] ---
    for (int nt = 0; nt < 2; nt++) {
      v8f acc =
          mm16(&t1A[wave][0][0], 36, &w_pe2p[0][0], 32, 32, nt * 16, m, half);
      float bv = b_pe2[nt * 16 + m];
#pragma unroll
      for (int r = 0; r < 8; r++) {
        float v = acc[r] + bv;
        catA[wave][r + 8 * half][32 + nt * 16 + m] = v > 0.f ? v : 0.f;
      }
    }
    __syncthreads();
    // --- ve = relu(cat @ fc_w + b) -> voxel max scatter ---
    for (int nt = 0; nt < 4; nt++) {
      v8f acc =
          mm16(&catA[wave][0][0], 68, &w_fcp[0][0], 64, 64, nt * 16, m, half);
      float bv = b_fc[nt * 16 + m];
#pragma unroll
      for (int r = 0; r < 8; r++) {
        int row = r + 8 * half;
        int s = segA[wave][row];
        if (s < NVOX) {  // valid point in a real voxel
          float v = acc[r] + bv;
          v = v > 0.f ? v : 0.f;
          // post-relu values >= 0: uint compare == float compare; 0 == empty
          atomicMax(&vox[(size_t)s * 64 + nt * 16 + m], __float_as_uint(v));
        }
      }
    }
    __syncthreads();  // staging buffers reused next iteration
  }
}

// ---------------- kernel 3: voxel attention + box scatter-add --------------
__global__ __launch_bounds__(256) void bfe_attn_kernel(
    const unsigned int* __restrict__ vox, const float* __restrict__ segstat,
    const float* __restrict__ grid_emb, const float* __restrict__ aw1,
    const float* __restrict__ ab1, const float* __restrict__ aw2,
    const float* __restrict__ ab2, float* __restrict__ box_feat, int NVOX) {
  __shared__ float w1[64][32];
  __shared__ float b1s[32];
  __shared__ float w2s[32];
  __shared__ float b2s;
  int tid = threadIdx.x;
  for (int i = tid; i < 64 * 32; i += 256) w1[i >> 5][i & 31] = aw1[i];
  if (tid < 32) {
    b1s[tid] = ab1[tid];
    w2s[tid] = aw2[tid];
  }
  if (tid == 0) b2s = ab2[0];
  __syncthreads();
  int v = blockIdx.x * 256 + tid;
  if (v >= NVOX) return;
  float cnt = segstat[(size_t)v * 4 + 3];
  if (cnt <= 0.f) return;  // empty voxels contribute nothing to box_feat
  float x[64];
  const float* ge = grid_emb + (size_t)(v % NCELL) * 64;
#pragma unroll
  for (int c = 0; c < 64; c++)
    x[c] = __uint_as_float(vox[(size_t)v * 64 + c]) + ge[c];
  float h[32];
#pragma unroll
  for (int j = 0; j < 32; j++) h[j] = b1s[j];
  for (int c = 0; c < 64; c++) {
    float xc = x[c];
#pragma unroll
    for (int j = 0; j < 32; j++) h[j] += xc * w1[c][j];
  }
  float z = b2s;
#pragma unroll
  for (int j = 0; j < 32; j++) z += (h[j] > 0.f ? h[j] : 0.f) * w2s[j];
  float a = 1.f / (1.f + expf(-z));
  float* bf = box_feat + (size_t)(v / NCELL) * 64;
  for (int c = 0; c < 64; c++) atomicAdd(&bf[c], a * x[c]);
}

// ---------------- kernel 4: output head ------------------------------------
__global__ void bfe_out_kernel(const float* __restrict__ box_feat,
                               const float* __restrict__ out_w,
                               const float* __restrict__ out_b,
                               float* __restrict__ out, int total) {
  int i = blockIdx.x * blockDim.x + threadIdx.x;
  if (i >= total) return;
  int n = i >> 6, o = i & 63;
  float acc = out_b[o];
  for (int c = 0; c < 64; c++)
    acc += box_feat[(size_t)n * 64 + c] * out_w[(size_t)c * 64 + o];
  out[i] = acc > 0.f ? acc : 0.f;
}

// ---------------- launcher --------------------------------------------------
extern "C" void kernel_launch(void* const* d_in, const int* in_sizes, int n_in,
                              void* d_out, int out_size, void* d_ws,
                              size_t ws_size, hipStream_t stream) {
  const float* coords = (const float*)d_in[0];
  const float* features = (const float*)d_in[1];
  const float* boxes = (const float*)d_in[2];
  const float* grid_emb = (const float*)d_in[3];
  const float* pe_w1 = (const float*)d_in[4];
  const float* pe_b1 = (const float*)d_in[5];
  const float* pe_w2 = (const float*)d_in[6];
  const float* pe_b2 = (const float*)d_in[7];
  const float* proj_w = (const float*)d_in[8];
  const float* proj_b = (const float*)d_in[9];
  const float* fc_w = (const float*)d_in[10];
  const float* fc_b = (const float*)d_in[11];
  const float* attn_w1 = (const float*)d_in[12];
  const float* attn_b1 = (const float*)d_in[13];
  const float* attn_w2 = (const float*)d_in[14];
  const float* attn_b2 = (const float*)d_in[15];
  const float* out_w = (const float*)d_in[16];
  const float* out_b = (const float*)d_in[17];

  const int P = in_sizes[0] / 4;
  const int N = in_sizes[2] / 8;
  const int NVOX = N * NCELL;
  const int S = NVOX + 1;

  // workspace layout (floats): segstat[S*4] | vox[NVOX*64] | box_feat[N*64]
  //                            | nxyz[P*4] | seg[P]
  float* ws = (float*)d_ws;
  float* segstat = ws;
  unsigned int* vox = (unsigned int*)(ws + (size_t)S * 4);
  float* box_feat = (float*)(vox + (size_t)NVOX * 64);
  float* nxyz = box_feat + (size_t)N * 64;
  int* seg = (int*)(nxyz + (size_t)P * 4);

  size_t zcount = (size_t)S * 4 + (size_t)NVOX * 64 + (size_t)N * 64;
  bfe_zero_kernel<<<(int)((zcount + 255) / 256), 256, 0, stream>>>(ws, zcount);

  bfe_pib_kernel<<<(P + 255) / 256, 256, 0, stream>>>(coords, boxes, nxyz, seg,
                                                      segstat, P, N);

  int nTiles = (P + 15) / 16;
  int tilesPerBlock = MLP_WAVES * TILES_PER_WAVE;
  bfe_mlp_kernel<<<(nTiles + tilesPerBlock - 1) / tilesPerBlock, 64, 0,
                   stream>>>(features, nxyz, seg, segstat, pe_w1, pe_b1, pe_w2,
                             pe_b2, proj_w, proj_b, fc_w, fc_b, vox, P, nTiles,
                             NVOX);

  bfe_attn_kernel<<<(NVOX + 255) / 256, 256, 0, stream>>>(
      vox, segstat, grid_emb, attn_w1, attn_b1, attn_w2, attn_b2, box_feat,
      NVOX);

  bfe_out_kernel<<<(N * 64 + 255) / 256, 256, 0, stream>>>(box_feat, out_w,
                                                           out_b, (float*)d_out,
                                                           N * 64);
}